// DetBenchPredict_41025527611447
// MI455X (gfx1250) — compile-verified
//
#include <hip/hip_runtime.h>
#include <hip/hip_bf16.h>

// ---------------------------------------------------------------------------
// EfficientDet post-process for MI455X (gfx1250, wave32).
// Roofline: 283 MB logit scan dominates -> ~12us/pass at 23.3 TB/s.
// No matmul FLOPs exist in this op, so WMMA is inapplicable; the CDNA5
// features used are the async-tensor-class memory path
// (global_load_async_to_lds_b128 + s_wait_asynccnt), global_prefetch_b8,
// and wave32 shuffle reductions. NMS runs entirely out of LDS (147KB/WG).
// ---------------------------------------------------------------------------

#define NUM_IMG   16
#define NUM_CLS   90
#define NUM_A     9
#define TOPK      5000
#define CAP       6144          // candidate cap per image (>= 5000 + one bin)
#define NBINS     1024
#define LOGIT_LO  (-16.0f)
#define BINS_PER_UNIT 32.0f     // bin width = 1/32, LO and width are pow2
#define NMS_ROUNDS 100

struct Cand {
  float *score, *cls;
  float *ox1, *oy1, *ox2, *oy2, *area;   // class-offset boxes (for NMS)
  float *bx1, *by1, *bx2, *by2;          // plain clamped boxes (for output)
};

// ---------------------------------------------------------------------------
// Kernel A: per-image logit histogram. Streams one level's slab per image
// through LDS with per-wave double-buffered async loads (ASYNCcnt path).
// ---------------------------------------------------------------------------
#define HIST_ITERS 16   // float4s per thread per block

__global__ __launch_bounds__(256) void hist_kernel(
    const float* __restrict__ cls, int f, int* __restrict__ hist)
{
  const int img = blockIdx.y;
  const long n  = (long)(NUM_A * NUM_CLS) * f * f;   // floats per image slab
  const long nv = n >> 2;                            // float4s per image slab
  const float4* src = (const float4*)(cls + (size_t)img * n);

  __shared__ float4 tile[2][256];
  __shared__ int hist_s[NBINS];
  for (int i = threadIdx.x; i < NBINS; i += blockDim.x) hist_s[i] = 0;
  __syncthreads();

  const int tid = threadIdx.x;
  // low 32 bits of a generic pointer to LDS == LDS byte offset (ISA 10.2)
  unsigned lds0 = (unsigned)(size_t)(&tile[0][tid]);
  unsigned lds1 = (unsigned)(size_t)(&tile[1][tid]);

  const long vbase = (long)blockIdx.x * (256L * HIST_ITERS) + tid;

  // prologue: issue first chunk into buffer 0 (OOB lanes clamp to slab start)
  {
    long v = vbase;
    const float4* p = src + (v < nv ? v : 0);
    asm volatile("global_load_async_to_lds_b128 %0, %1, off"
                 :: "v"(lds0), "v"(p) : "memory");
  }

  for (int it = 0; it < HIST_ITERS; ++it) {
    const int cur = it & 1;
    if (it + 1 < HIST_ITERS) {
      long vn = vbase + (long)(it + 1) * 256L;
      const float4* p = src + (vn < nv ? vn : 0);
      unsigned ldsn = (cur == 0) ? lds1 : lds0;
      asm volatile("global_load_async_to_lds_b128 %0, %1, off"
                   :: "v"(ldsn), "v"(p) : "memory");
      asm volatile("s_wait_asynccnt 0x1" ::: "memory");  // prev chunk done
    } else {
      asm volatile("s_wait_asynccnt 0x0" ::: "memory");  // last chunk done
    }
    long vc = vbase + (long)it * 256L;
    if (vc < nv) {
      float4 d = tile[cur][tid];
      float vals[4] = {d.x, d.y, d.z, d.w};
      #pragma unroll
      for (int k = 0; k < 4; ++k) {
        int b = (int)floorf((vals[k] - LOGIT_LO) * BINS_PER_UNIT);
        b = b < 0 ? 0 : (b > NBINS - 1 ? NBINS - 1 : b);
        atomicAdd(&hist_s[b], 1);
      }
    }
  }
  __syncthreads();
  for (int b = threadIdx.x; b < NBINS; b += blockDim.x) {
    int c = hist_s[b];
    if (c) atomicAdd(&hist[img * NBINS + b], c);
  }
}

// ---------------------------------------------------------------------------
// Kernel B: walk histogram top-down to the top-5000 threshold per image.
// ---------------------------------------------------------------------------
__global__ void thresh_kernel(const int* __restrict__ hist,
                              float* __restrict__ thresh)
{
  int img = threadIdx.x;
  if (img >= NUM_IMG) return;
  long cum = 0;
  float th = LOGIT_LO;
  for (int b = NBINS - 1; b >= 0; --b) {
    cum += hist[img * NBINS + b];
    if (cum >= TOPK) { th = LOGIT_LO + (float)b * (1.0f / BINS_PER_UNIT); break; }
  }
  thresh[img] = th;
}

// ---------------------------------------------------------------------------
// Kernel C: second streaming pass — compact logits >= threshold and decode
// boxes (anchor decode, clamp, sigmoid, class offset) in one shot.
// ---------------------------------------------------------------------------
__global__ __launch_bounds__(256) void select_decode_kernel(
    const float* __restrict__ cls, const float* __restrict__ box,
    const float* __restrict__ anchors,
    const float* __restrict__ scales, const float* __restrict__ sizes,
    int f, int ancBase,
    const float* __restrict__ thresh, int* __restrict__ count, Cand cd)
{
  const int img = blockIdx.y;
  const int ff  = f * f;
  const int n   = NUM_A * NUM_CLS * ff;
  const float* src = cls + (size_t)img * n;
  const float th    = thresh[img];
  const float scale = scales[img];
  const float limx  = sizes[img * 2 + 0] / scale;
  const float limy  = sizes[img * 2 + 1] / scale;
  const float sep   = fmaxf(limx, limy) + 1.0f;   // class separator (>= max box)
  const int stride  = gridDim.x * blockDim.x;

  for (int e = blockIdx.x * blockDim.x + threadIdx.x; e < n; e += stride) {
    if (e + stride < n) __builtin_prefetch(src + e + stride, 0, 1); // global_prefetch_b8
    float logit = src[e];
    if (logit >= th) {
      int slot = atomicAdd(&count[img], 1);
      if (slot < CAP) {
        int ch  = e / ff;  int rem = e - ch * ff;
        int h   = rem / f; int w   = rem - h * f;
        int a   = ch / NUM_CLS;
        int c   = ch - a * NUM_CLS;
        int gpos = ancBase + (h * f + w) * NUM_A + a;
        const float* anc = anchors + (size_t)gpos * 4;
        float a0 = anc[0], a1 = anc[1], a2 = anc[2], a3 = anc[3];
        float yca = (a0 + a2) * 0.5f, xca = (a1 + a3) * 0.5f;
        float ha  = a2 - a0,          wa  = a3 - a1;
        const float* bp = box + ((size_t)img * (NUM_A * 4) + a * 4) * ff + rem;
        float ty = bp[0], tx = bp[ff], thh = bp[2 * ff], tw = bp[3 * ff];
        float bw = expf(tw) * wa, bh = expf(thh) * ha;
        float yc = ty * ha + yca, xc = tx * wa + xca;
        float x1 = xc - bw * 0.5f, y1 = yc - bh * 0.5f;
        float x2 = xc + bw * 0.5f, y2 = yc + bh * 0.5f;
        x1 = fminf(fmaxf(x1, 0.f), limx); x2 = fminf(fmaxf(x2, 0.f), limx);
        y1 = fminf(fmaxf(y1, 0.f), limy); y2 = fminf(fmaxf(y2, 0.f), limy);
        float sc  = 1.0f / (1.0f + expf(-logit));
        float off = (float)c * sep;
        size_t s = (size_t)img * CAP + slot;
        cd.score[s] = sc;       cd.cls[s] = (float)c;
        cd.bx1[s] = x1;         cd.by1[s] = y1;
        cd.bx2[s] = x2;         cd.by2[s] = y2;
        cd.ox1[s] = x1 + off;   cd.oy1[s] = y1 + off;
        cd.ox2[s] = x2 + off;   cd.oy2[s] = y2 + off;
        cd.area[s] = (x2 - x1) * (y2 - y1);
      }
    }
  }
}

// ---------------------------------------------------------------------------
// Kernel D: 100-round NMS per image, one workgroup per image, candidates
// held in 147KB of dynamic LDS, wave32 shuffle argmax reduction.
// ---------------------------------------------------------------------------
__global__ __launch_bounds__(256) void nms_kernel(
    Cand cd, const int* __restrict__ count,
    const float* __restrict__ scales, float* __restrict__ out)
{
  const int img = blockIdx.x;
  extern __shared__ float smem[];
  float* sx1 = smem;
  float* sy1 = sx1 + CAP;
  float* sx2 = sy1 + CAP;
  float* sy2 = sx2 + CAP;
  float* sar = sy2 + CAP;
  float* ssc = sar + CAP;      // working score; -1 == suppressed/invalid

  __shared__ float wbest[8];
  __shared__ int   widx[8];
  __shared__ float chS;
  __shared__ int   chI;
  __shared__ int   keepI[NMS_ROUNDS];
  __shared__ unsigned char keepOk[NMS_ROUNDS];

  int n = count[img];
  if (n > CAP) n = CAP;
  const size_t cb = (size_t)img * CAP;

  for (int i = threadIdx.x; i < CAP; i += blockDim.x) {
    if (i < n) {
      sx1[i] = cd.ox1[cb + i]; sy1[i] = cd.oy1[cb + i];
      sx2[i] = cd.ox2[cb + i]; sy2[i] = cd.oy2[cb + i];
      sar[i] = cd.area[cb + i];
      ssc[i] = cd.score[cb + i];
    } else {
      ssc[i] = -1.0f;
    }
  }
  __syncthreads();

  for (int r = 0; r < NMS_ROUNDS; ++r) {
    // ---- parallel argmax over LDS scores ----
    float bs = -1.0f; int bi = 0;
    for (int i = threadIdx.x; i < CAP; i += blockDim.x) {
      float s = ssc[i];
      if (s > bs || (s == bs && i < bi)) { bs = s; bi = i; }
    }
    #pragma unroll
    for (int o = 16; o > 0; o >>= 1) {       // wave32 butterfly
      float os = __shfl_down(bs, o, 32);
      int   oi = __shfl_down(bi, o, 32);
      if (os > bs || (os == bs && oi < bi)) { bs = os; bi = oi; }
    }
    const int lane = threadIdx.x & 31, wv = threadIdx.x >> 5;
    if (lane == 0) { wbest[wv] = bs; widx[wv] = bi; }
    __syncthreads();
    if (threadIdx.x == 0) {
      float s0 = wbest[0]; int i0 = widx[0];
      for (int w2 = 1; w2 < 8; ++w2) {
        float s = wbest[w2]; int ii = widx[w2];
        if (s > s0 || (s == s0 && ii < i0)) { s0 = s; i0 = ii; }
      }
      chS = s0; chI = i0;
      keepI[r]  = i0;
      keepOk[r] = (s0 > -0.5f) ? 1 : 0;
      ssc[i0] = -1.0f;
    }
    __syncthreads();

    // ---- suppress by IoU against the chosen box ----
    const int ci = chI;
    const float cx1 = sx1[ci], cy1 = sy1[ci], cx2 = sx2[ci], cy2 = sy2[ci];
    const float ca  = (cx2 - cx1) * (cy2 - cy1);
    for (int i = threadIdx.x; i < CAP; i += blockDim.x) {
      if (ssc[i] > -0.5f) {
        float iw = fminf(sx2[i], cx2) - fmaxf(sx1[i], cx1);
        float ih = fminf(sy2[i], cy2) - fmaxf(sy1[i], cy1);
        float inter = fmaxf(iw, 0.f) * fmaxf(ih, 0.f);
        float iou = inter / (sar[i] + ca - inter);
        if (!(iou <= 0.5f)) ssc[i] = -1.0f;   // NaN also suppresses (ref behavior)
      }
    }
    __syncthreads();
  }

  // ---- write detections (16,100,6): [x, y, w, h, score, class+1] ----
  const float scale = scales[img];
  if (threadIdx.x < NMS_ROUNDS) {
    const int r = threadIdx.x;
    float* o = out + ((size_t)img * NMS_ROUNDS + r) * 6;
    if (keepOk[r]) {
      const int ci = keepI[r];
      float x1 = cd.bx1[cb + ci], y1 = cd.by1[cb + ci];
      float x2 = cd.bx2[cb + ci], y2 = cd.by2[cb + ci];
      o[0] = x1 * scale;        o[1] = y1 * scale;
      o[2] = (x2 - x1) * scale; o[3] = (y2 - y1) * scale;
      o[4] = cd.score[cb + ci]; o[5] = cd.cls[cb + ci] + 1.0f;
    } else {
      o[0] = 0.f; o[1] = 0.f; o[2] = 0.f; o[3] = 0.f; o[4] = 0.f; o[5] = 0.f;
    }
  }
}

// ---------------------------------------------------------------------------
// Host launcher
// ---------------------------------------------------------------------------
extern "C" void kernel_launch(void* const* d_in, const int* in_sizes, int n_in,
                              void* d_out, int out_size, void* d_ws, size_t ws_size,
                              hipStream_t stream) {
  (void)in_sizes; (void)n_in; (void)out_size; (void)ws_size;

  // setup_inputs() dict order: cls_l3, box_l3, cls_l4, box_l4, ... anchors, scales, size
  const float* cls_l[5]; const float* box_l[5];
  for (int l = 0; l < 5; ++l) {
    cls_l[l] = (const float*)d_in[2 * l + 0];
    box_l[l] = (const float*)d_in[2 * l + 1];
  }
  const float* anchors = (const float*)d_in[10];
  const float* scales  = (const float*)d_in[11];
  const float* sizes   = (const float*)d_in[12];
  float* out = (float*)d_out;

  static const int feats[5]    = {64, 32, 16, 8, 4};
  static const int ancBase[5]  = {0, 36864, 46080, 48384, 48960};

  // workspace layout
  int*   hist   = (int*)d_ws;                      // 16*1024
  int*   count  = hist + NUM_IMG * NBINS;          // 16
  float* thresh = (float*)(count + NUM_IMG);       // 16
  float* cbase  = thresh + NUM_IMG;
  Cand cd;
  cd.score = cbase + 0  * (size_t)NUM_IMG * CAP;
  cd.cls   = cbase + 1  * (size_t)NUM_IMG * CAP;
  cd.ox1   = cbase + 2  * (size_t)NUM_IMG * CAP;
  cd.oy1   = cbase + 3  * (size_t)NUM_IMG * CAP;
  cd.ox2   = cbase + 4  * (size_t)NUM_IMG * CAP;
  cd.oy2   = cbase + 5  * (size_t)NUM_IMG * CAP;
  cd.area  = cbase + 6  * (size_t)NUM_IMG * CAP;
  cd.bx1   = cbase + 7  * (size_t)NUM_IMG * CAP;
  cd.by1   = cbase + 8  * (size_t)NUM_IMG * CAP;
  cd.bx2   = cbase + 9  * (size_t)NUM_IMG * CAP;
  cd.by2   = cbase + 10 * (size_t)NUM_IMG * CAP;

  // zero histograms + counters (d_ws is poisoned; must re-init every call)
  hipMemsetAsync(d_ws, 0, (size_t)(NUM_IMG * NBINS + NUM_IMG) * sizeof(int), stream);

  // Pass A: histogram (async-LDS streaming)
  for (int l = 0; l < 5; ++l) {
    const int f  = feats[l];
    const long n = (long)(NUM_A * NUM_CLS) * f * f;
    const long nv = n >> 2;
    const int bx = (int)((nv + 256L * HIST_ITERS - 1) / (256L * HIST_ITERS));
    hist_kernel<<<dim3(bx, NUM_IMG, 1), 256, 0, stream>>>(cls_l[l], f, hist);
  }

  // Pass B: threshold
  thresh_kernel<<<1, NUM_IMG, 0, stream>>>(hist, thresh);

  // Pass C: compact + decode
  for (int l = 0; l < 5; ++l) {
    const int f = feats[l];
    const int n = NUM_A * NUM_CLS * f * f;
    int bx = (n + 255) / 256; if (bx > 1024) bx = 1024;
    select_decode_kernel<<<dim3(bx, NUM_IMG, 1), 256, 0, stream>>>(
        cls_l[l], box_l[l], anchors, scales, sizes, f, ancBase[l],
        thresh, count, cd);
  }

  // Pass D: NMS out of LDS (6 arrays * 6144 * 4B = 147456 B dynamic LDS)
  const size_t smem = (size_t)6 * CAP * sizeof(float);
  nms_kernel<<<NUM_IMG, 256, smem, stream>>>(cd, count, scales, out);
}